// CodexNetwork_12463995093744
// MI455X (gfx1250) — compile-verified
//
#include <hip/hip_runtime.h>
#include <math.h>

// ---------------------------------------------------------------------------
// Shapes (fixed by the reference): B=8, N=512, D=256, R=6, HEADS=3
//   d_out layout (f32, flat, return order):
//     [0 .. 1048576)            output   [B,N,D]
//     [1048576 .. +3*4096)      interp   heads 0..2, each [B,N,1]
//     [1060864 .. +1048576)     symbols  [B,N,D]   (post positional grammar)
//   d_ws layout (f32):
//     symbols(embed)  : 1,048,576
//     T[b,r,i,e]      : 6,291,456   (= symbols @ rel_W per relation)
//     cnt[b,j,r]      :    24,576
//     h[r,m,e]        : 3,145,728   (gelu hidden of the 3 heads)
//   total ws ~42 MB  (fits easily in the 192 MB L2)
// ---------------------------------------------------------------------------

typedef float v2f __attribute__((ext_vector_type(2)));
typedef float v8f __attribute__((ext_vector_type(8)));

#define TPB 256   // 8 wave32 waves
#define MB  64    // block tile M
#define NB  32    // block tile N
#define KB  16    // K stage depth (4 wmma k-steps)
#define SA  18    // LDS A row stride (pad vs bank conflicts, keeps b64 align)
#define SB  33    // LDS B row stride

__device__ __forceinline__ int relType(float dx, float dy) {
  // jnp.select first-true-wins; dx = x_j - x_i, dy = y_j - y_i
  if (dy >  0.5f) return 0;
  if (dy < -0.5f) return 1;
  if (dx < -0.5f) return 2;
  if (dx >  0.5f) return 3;
  if (fabsf(dx) < 0.3f && fabsf(dy) < 0.3f) return 4;
  return 5;
}

__device__ __forceinline__ float gelu_exact(float x) {
  return 0.5f * x * (1.0f + erff(x * 0.70710678118654752440f));
}

// Shared 64x32 block-tile GEMM core, f32 WMMA 16x16x4, K multiple of 16,
// N multiple of 32, A row-major [*,K], B row-major [K,N]. Returns this
// wave's 16x16 accumulator (wave tile: mw = wid&3, nw = wid>>2).
__device__ __forceinline__ v8f gemm_tile_core(
    const float* __restrict__ A, const float* __restrict__ Bm,
    int K, int N, int m0, int n0, float* As, float* Bs)
{
  const int tid  = threadIdx.x;
  const int lane = tid & 31, wid = tid >> 5;
  const int mw = wid & 3, nw = wid >> 2;
  const int half = lane >> 4, mr = lane & 15;
  v8f acc = {};
  for (int k0 = 0; k0 < K; k0 += KB) {
    { // stage A tile 64x16 (coalesced float4 per thread)
      const int r = tid >> 2, c4 = (tid & 3) * 4;
      const float4 v = *(const float4*)(A + (size_t)(m0 + r) * K + k0 + c4);
      float* dst = As + r * SA + c4;
      dst[0] = v.x; dst[1] = v.y; dst[2] = v.z; dst[3] = v.w;
      if (k0 + KB < K)  // emits global_prefetch for next stage
        __builtin_prefetch(A + (size_t)(m0 + r) * K + k0 + KB + c4, 0, 1);
    }
    { // stage B tile 16x32
      const int r = tid >> 4, c = (tid & 15) * 2;
      const float2 v = *(const float2*)(Bm + (size_t)(k0 + r) * N + n0 + c);
      Bs[r * SB + c] = v.x; Bs[r * SB + c + 1] = v.y;
      if (k0 + KB < K)
        __builtin_prefetch(Bm + (size_t)(k0 + KB + r) * N + n0 + c, 0, 1);
    }
    __syncthreads();
#pragma unroll
    for (int kk = 0; kk < 4; ++kk) {
      // ISA 16x16x4 f32 layouts: A lanes0-15 M=0..15; K = 2*half + vgpr.
      const int kl = kk * 4 + 2 * half;
      v2f a, b;
      const float* ap = As + (mw * 16 + mr) * SA + kl;
      a.x = ap[0];
      a.y = ap[1];
      const int nn = nw * 16 + mr;
      b.x = Bs[kl * SB + nn];
      b.y = Bs[(kl + 1) * SB + nn];
      acc = __builtin_amdgcn_wmma_f32_16x16x4_f32(
          false, a, false, b, (short)0, acc, false, false);
    }
    __syncthreads();
  }
  return acc;
}

// ---------------- kernel 1: embedding gather ----------------
__global__ __launch_bounds__(TPB) void k_embed(
    const int* __restrict__ ids, const float* __restrict__ sym,
    const float* __restrict__ lay, float* __restrict__ out)
{
  const int idx = blockIdx.x * TPB + threadIdx.x;     // < 8192*TPB = 2,097,152 (= 2*B*N*D)
  const int bn = idx >> 8, d = idx & 255;
  const int sid = ids[bn];
  out[idx] = sym[sid * 256 + d] + lay[sid * 256 + d];
}

// ---------------- kernel 2: per-(b,j) relation counts ----------------
__global__ __launch_bounds__(TPB) void k_counts(
    const float* __restrict__ positions, float* __restrict__ cnt)
{
  __shared__ float px[512], py[512];
  const int b = blockIdx.x;
  const int j = blockIdx.y * TPB + threadIdx.x;
  for (int i = threadIdx.x; i < 512; i += TPB) {
    px[i] = positions[((size_t)b * 512 + i) * 2 + 0];
    py[i] = positions[((size_t)b * 512 + i) * 2 + 1];
  }
  __syncthreads();
  const float xj = px[j], yj = py[j];
  float c0 = 0, c1 = 0, c2 = 0, c3 = 0, c4 = 0, c5 = 0;
  for (int i = 0; i < 512; ++i) {
    const int r = relType(xj - px[i], yj - py[i]);
    const float ok = (i != j) ? 1.0f : 0.0f;
    c0 += ok * (float)(r == 0); c1 += ok * (float)(r == 1);
    c2 += ok * (float)(r == 2); c3 += ok * (float)(r == 3);
    c4 += ok * (float)(r == 4); c5 += ok * (float)(r == 5);
  }
  float* o = cnt + ((size_t)b * 512 + j) * 6;
  o[0] = c0; o[1] = c1; o[2] = c2; o[3] = c3; o[4] = c4; o[5] = c5;
}

// ---------------- kernel 3: T[b,r] = symbols[b] @ rel_W[r] ----------------
__global__ __launch_bounds__(TPB) void k_gemm_T(
    const float* __restrict__ symbols, const float* __restrict__ rel_W,
    float* __restrict__ T)
{
  __shared__ float As[MB * SA];
  __shared__ float Bs[KB * SB];
  const int z = blockIdx.z, b = z / 6, r = z % 6;
  const int m0 = blockIdx.y * MB, n0 = blockIdx.x * NB;
  const v8f acc = gemm_tile_core(symbols + (size_t)b * 512 * 256,
                                 rel_W + (size_t)r * 256 * 256,
                                 256, 256, m0, n0, As, Bs);
  const int lane = threadIdx.x & 31, wid = threadIdx.x >> 5;
  const int mw = wid & 3, nw = wid >> 2, half = lane >> 4, mr = lane & 15;
  float* D = T + (size_t)z * 512 * 256;
  const int n = n0 + nw * 16 + mr;
#pragma unroll
  for (int v = 0; v < 8; ++v) {
    const int m = m0 + mw * 16 + v + 8 * half;   // C/D layout: m = vgpr + 8*half
    D[(size_t)m * 256 + n] = acc[v];
  }
}

// ------- kernel 4: mask GEMM pooling: symOut = M @ T + symbols + cnt.rel_b -------
// Per batch b: A[j, k=(r*512+i)] = (rel(i,j)==r && i!=j) ? 1 : 0, generated
// on the fly from positions in LDS; B = T[b] viewed [3072, 256].
__global__ __launch_bounds__(TPB) void k_gemm_pool(
    const float* __restrict__ T, const float* __restrict__ positions,
    const float* __restrict__ symbols, const float* __restrict__ cnt,
    const float* __restrict__ rel_b, float* __restrict__ symOut)
{
  __shared__ float Bs[KB * SB];
  __shared__ float px[512], py[512];
  const int b = blockIdx.z;
  const int m0 = blockIdx.y * MB, n0 = blockIdx.x * NB;
  const int tid = threadIdx.x, lane = tid & 31, wid = tid >> 5;
  const int mw = wid & 3, nw = wid >> 2, half = lane >> 4, mr = lane & 15;
  for (int i = tid; i < 512; i += TPB) {
    px[i] = positions[((size_t)b * 512 + i) * 2 + 0];
    py[i] = positions[((size_t)b * 512 + i) * 2 + 1];
  }
  __syncthreads();
  const int j = m0 + mw * 16 + mr;         // this lane's A-matrix row (target)
  const float xj = px[j], yj = py[j];
  const float* Bm = T + (size_t)b * 3072 * 256;
  v8f acc = {};
  for (int k0 = 0; k0 < 3072; k0 += KB) {
    {
      const int rr = tid >> 4, c = (tid & 15) * 2;
      const float2 v = *(const float2*)(Bm + (size_t)(k0 + rr) * 256 + n0 + c);
      Bs[rr * SB + c] = v.x; Bs[rr * SB + c + 1] = v.y;
      if (k0 + KB < 3072)
        __builtin_prefetch(Bm + (size_t)(k0 + KB + rr) * 256 + n0 + c, 0, 1);
    }
    __syncthreads();
#pragma unroll
    for (int kk = 0; kk < 4; ++kk) {
      const int kb = k0 + kk * 4;
      const int r  = kb >> 9;               // relation index (k0 never straddles)
      const int i0 = (kb & 511) + 2 * half; // source symbol for a.x; a.y is i0+1
      v2f a, bf;
      a.x = (i0     != j && relType(xj - px[i0],     yj - py[i0])     == r) ? 1.0f : 0.0f;
      a.y = (i0 + 1 != j && relType(xj - px[i0 + 1], yj - py[i0 + 1]) == r) ? 1.0f : 0.0f;
      const int kl = kk * 4 + 2 * half;
      const int nn = nw * 16 + mr;
      bf.x = Bs[kl * SB + nn];
      bf.y = Bs[(kl + 1) * SB + nn];
      acc = __builtin_amdgcn_wmma_f32_16x16x4_f32(
          false, a, false, bf, (short)0, acc, false, false);
    }
    __syncthreads();
  }
  // epilogue: + residual symbols + per-relation count bias
  const int n = n0 + nw * 16 + mr;
  float rb[6];
#pragma unroll
  for (int r = 0; r < 6; ++r) rb[r] = rel_b[r * 256 + n];
#pragma unroll
  for (int v = 0; v < 8; ++v) {
    const int jm = m0 + mw * 16 + v + 8 * half;
    const size_t row = (size_t)b * 512 + jm;
    float val = acc[v] + symbols[row * 256 + n];
#pragma unroll
    for (int r = 0; r < 6; ++r) val = fmaf(cnt[row * 6 + r], rb[r], val);
    symOut[row * 256 + n] = val;
  }
}

// ------- kernel 5: h[r] = gelu(symOut @ h1_W[r] + h1_b[r]) -------
__global__ __launch_bounds__(TPB) void k_gemm_heads(
    const float* __restrict__ symOut, const float* __restrict__ h1_W,
    const float* __restrict__ h1_b, float* __restrict__ h)
{
  __shared__ float As[MB * SA];
  __shared__ float Bs[KB * SB];
  const int z = blockIdx.z;                 // head 0..2
  const int m0 = blockIdx.y * MB, n0 = blockIdx.x * NB;
  const v8f acc = gemm_tile_core(symOut, h1_W + (size_t)z * 256 * 256,
                                 256, 256, m0, n0, As, Bs);
  const int lane = threadIdx.x & 31, wid = threadIdx.x >> 5;
  const int mw = wid & 3, nw = wid >> 2, half = lane >> 4, mr = lane & 15;
  const int n = n0 + nw * 16 + mr;
  const float bias = h1_b[z * 256 + n];
  float* D = h + (size_t)z * 4096 * 256;
#pragma unroll
  for (int v = 0; v < 8; ++v) {
    const int m = m0 + mw * 16 + v + 8 * half;
    D[(size_t)m * 256 + n] = gelu_exact(acc[v] + bias);
  }
}

// ------- kernel 6: output = symOut @ out_W + out_b -------
__global__ __launch_bounds__(TPB) void k_gemm_out(
    const float* __restrict__ symOut, const float* __restrict__ out_W,
    const float* __restrict__ out_b, float* __restrict__ out)
{
  __shared__ float As[MB * SA];
  __shared__ float Bs[KB * SB];
  const int m0 = blockIdx.y * MB, n0 = blockIdx.x * NB;
  const v8f acc = gemm_tile_core(symOut, out_W, 256, 256, m0, n0, As, Bs);
  const int lane = threadIdx.x & 31, wid = threadIdx.x >> 5;
  const int mw = wid & 3, nw = wid >> 2, half = lane >> 4, mr = lane & 15;
  const int n = n0 + nw * 16 + mr;
  const float bias = out_b[n];
#pragma unroll
  for (int v = 0; v < 8; ++v) {
    const int m = m0 + mw * 16 + v + 8 * half;
    out[(size_t)m * 256 + n] = acc[v] + bias;
  }
}

// ------- kernel 7: interp[r,m] = h[r,m,:] . h2_W[r,:] + h2_b[r] -------
__global__ __launch_bounds__(TPB) void k_interp(
    const float* __restrict__ h, const float* __restrict__ h2_W,
    const float* __restrict__ h2_b, float* __restrict__ out)
{
  const int wid = threadIdx.x >> 5, lane = threadIdx.x & 31;
  const int row = blockIdx.x * 8 + wid;     // 0 .. 3*4096-1
  const int r = row >> 12, m = row & 4095;
  const float* hp = h + (size_t)row * 256;
  const float* wp = h2_W + r * 256;
  float s = 0.0f;
#pragma unroll
  for (int t = 0; t < 8; ++t) {
    const int e = lane + t * 32;
    s = fmaf(hp[e], wp[e], s);
  }
  for (int off = 16; off > 0; off >>= 1) s += __shfl_xor(s, off, 32);
  if (lane == 0) out[1048576 + r * 4096 + m] = s + h2_b[r];
}

// ---------------------------------------------------------------------------
extern "C" void kernel_launch(void* const* d_in, const int* in_sizes, int n_in,
                              void* d_out, int out_size, void* d_ws, size_t ws_size,
                              hipStream_t stream) {
  (void)in_sizes; (void)n_in; (void)out_size; (void)ws_size;
  const int*   symbol_ids = (const int*)d_in[0];
  const float* positions  = (const float*)d_in[1];
  const float* sym_table  = (const float*)d_in[2];
  const float* layer_tab  = (const float*)d_in[3];
  const float* rel_W      = (const float*)d_in[4];
  const float* rel_b      = (const float*)d_in[5];
  const float* h1_W       = (const float*)d_in[6];
  const float* h1_b       = (const float*)d_in[7];
  const float* h2_W       = (const float*)d_in[8];
  const float* h2_b       = (const float*)d_in[9];
  const float* out_W      = (const float*)d_in[10];
  const float* out_b      = (const float*)d_in[11];

  float* out = (float*)d_out;
  float* ws  = (float*)d_ws;
  float* symbols = ws;                                   // 1,048,576
  float* T       = ws + 1048576;                         // 6,291,456
  float* cnt     = ws + 1048576 + 6291456;               //    24,576
  float* h       = ws + 1048576 + 6291456 + 24576;       // 3,145,728
  float* symOut  = out + 1060864;                        // symbols output region

  k_embed     <<<dim3(8192),        TPB, 0, stream>>>(symbol_ids, sym_table, layer_tab, symbols);
  k_counts    <<<dim3(8, 2),        TPB, 0, stream>>>(positions, cnt);
  k_gemm_T    <<<dim3(8, 8, 48),    TPB, 0, stream>>>(symbols, rel_W, T);
  k_gemm_pool <<<dim3(8, 8, 8),     TPB, 0, stream>>>(T, positions, symbols, cnt, rel_b, symOut);
  k_gemm_heads<<<dim3(8, 64, 3),    TPB, 0, stream>>>(symOut, h1_W, h1_b, h);
  k_gemm_out  <<<dim3(8, 64),       TPB, 0, stream>>>(symOut, out_W, out_b, out);
  k_interp    <<<dim3(1536),        TPB, 0, stream>>>(h, h2_W, h2_b, out);
}